// EgoContextExtractor_40716289966164
// MI455X (gfx1250) — compile-verified
//
#include <hip/hip_runtime.h>
#include <hip/hip_bf16.h>
#include <math.h>

// Problem constants (match the reference).
#define BB 8
#define NN 512
#define HH 128
#define EE 130816      // N*(N-1)/2
#define ROWS (BB * NN) // 4096 node rows

typedef float v2f __attribute__((ext_vector_type(2)));
typedef float v4f __attribute__((ext_vector_type(4)));
typedef float v8f __attribute__((ext_vector_type(8)));

// ---------------------------------------------------------------------------
// Kernel 1: node GEMM via V_WMMA_F32_16X16X4_F32.
//   A[r,o] = sum_h X[r,h]*W1[o,h] + (b1[o]+b2[o])
//   C[r,o] = sum_h X[r,h]*(W2[o,h]-W1[o,h])
// One wave computes one 16x16 tile of both A and C (shared X fragment).
// Tiles: 256 row-tiles x 8 col-tiles = 2048 waves.
// ---------------------------------------------------------------------------
__global__ __launch_bounds__(256)
void node_gemm_kernel(const float* __restrict__ X,
                      const float* __restrict__ W1,
                      const float* __restrict__ b1,
                      const float* __restrict__ W2,
                      const float* __restrict__ b2,
                      float* __restrict__ Aout,
                      float* __restrict__ Cout) {
    const int wid  = blockIdx.x * (blockDim.x >> 5) + (threadIdx.x >> 5);
    const int rt   = wid >> 3;          // row tile 0..255
    const int ct   = wid & 7;           // col tile 0..7
    const int lane = threadIdx.x & 31;
    const int g    = lane >> 4;         // half-wave group (0/1)
    const int n    = lane & 15;         // column / M-row within tile

    // A-fragment row: M = lane % 16 for both halves.  K pair at k0 + 2*g.
    const float* xrow  = X  + (size_t)(rt * 16 + n) * HH;
    // B-fragment: column o = ct*16 + n, K pair at k0 + 2*g (row of W).
    const float* w1row = W1 + (size_t)(ct * 16 + n) * HH;
    const float* w2row = W2 + (size_t)(ct * 16 + n) * HH;

    v8f acc1 = {};  // X * W1^T
    v8f acc2 = {};  // X * (W2-W1)^T

#pragma unroll 4
    for (int k0 = 0; k0 < HH; k0 += 4) {
        const int kk = k0 + 2 * g;
        v2f afrag  = *(const v2f*)(xrow  + kk);
        v2f bfrag1 = *(const v2f*)(w1row + kk);
        v2f bfrag2 = *(const v2f*)(w2row + kk);
        v2f bdiff  = bfrag2 - bfrag1;           // W2 - W1

        acc1 = __builtin_amdgcn_wmma_f32_16x16x4_f32(
            /*neg_a=*/false, afrag, /*neg_b=*/false, bfrag1,
            /*c_mod=*/(short)0, acc1, /*reuse_a=*/false, /*reuse_b=*/false);
        acc2 = __builtin_amdgcn_wmma_f32_16x16x4_f32(
            /*neg_a=*/false, afrag, /*neg_b=*/false, bdiff,
            /*c_mod=*/(short)0, acc2, /*reuse_a=*/false, /*reuse_b=*/false);
    }

    // Bias (b1+b2) folded into A; bias depends only on output column o.
    const int o = ct * 16 + n;
    const float biasv = b1[o] + b2[o];

    // D layout: VGPR v holds row M = v + 8*g, column n.
#pragma unroll
    for (int v = 0; v < 8; ++v) {
        const int row = rt * 16 + v + 8 * g;
        Aout[(size_t)row * HH + o] = acc1[v] + biasv;
        Cout[(size_t)row * HH + o] = acc2[v];
    }
}

// ---------------------------------------------------------------------------
// Kernel 2: edge gather + add + relu, streaming 536 MB out.
// One wave per (b, e); lane L handles floats [4L, 4L+4) of the 128-float row.
// (i, j) recovered from e via reversed-triangle closed form.
// ---------------------------------------------------------------------------
__global__ __launch_bounds__(256)
void edge_kernel(const float* __restrict__ A,
                 const float* __restrict__ C,
                 float* __restrict__ out) {
    const int wave = threadIdx.x >> 5;
    const int lane = threadIdx.x & 31;
    const int e    = blockIdx.x * 8 + wave;   // 0..E-1 (E % 8 == 0)
    const int b    = blockIdx.y;              // 0..B-1

    // Reverse index: ep counts from the last edge; row-from-end t satisfies
    // t(t+1)/2 <= ep < (t+1)(t+2)/2.
    const unsigned ep = (unsigned)(EE - 1 - e);
    int t = (int)(0.5f * (sqrtf((float)(8u * ep + 1u)) - 1.0f));
    while ((unsigned)((t + 1) * (t + 2) / 2) <= ep) ++t;
    while (t > 0 && (unsigned)(t * (t + 1) / 2) > ep) --t;
    const int i = NN - 2 - t;
    const int j = NN - 1 - (int)(ep - (unsigned)(t * (t + 1) / 2));

    const int h = lane * 4;
    const v4f va = *(const v4f*)(A + ((size_t)(b * NN + i) * HH) + h);
    const v4f vc = *(const v4f*)(C + ((size_t)(b * NN + j) * HH) + h);

    v4f r = va + vc;
    r.x = fmaxf(r.x, 0.0f);
    r.y = fmaxf(r.y, 0.0f);
    r.z = fmaxf(r.z, 0.0f);
    r.w = fmaxf(r.w, 0.0f);

    const size_t off = ((size_t)b * EE + (size_t)e) * HH + h;
    __builtin_nontemporal_store(r, (v4f*)(out + off));
}

// ---------------------------------------------------------------------------
extern "C" void kernel_launch(void* const* d_in, const int* in_sizes, int n_in,
                              void* d_out, int out_size, void* d_ws, size_t ws_size,
                              hipStream_t stream) {
    const float* X  = (const float*)d_in[0];
    const float* W1 = (const float*)d_in[1];
    const float* b1 = (const float*)d_in[2];
    const float* W2 = (const float*)d_in[3];
    const float* b2 = (const float*)d_in[4];
    float* out = (float*)d_out;

    float* Abuf = (float*)d_ws;                      // 4096*128 floats (2 MB)
    float* Cbuf = Abuf + (size_t)ROWS * HH;          // 4096*128 floats (2 MB)

    // Stage 1: node GEMMs (2048 waves, 8 waves/block -> 256 blocks).
    node_gemm_kernel<<<dim3(256), dim3(256), 0, stream>>>(X, W1, b1, W2, b2,
                                                          Abuf, Cbuf);

    // Stage 2: edge assembly (B*E waves; E/8 = 16352 blocks in x, B in y).
    edge_kernel<<<dim3(EE / 8, BB), dim3(256), 0, stream>>>(Abuf, Cbuf, out);
}